// GCN_15290083573781
// MI455X (gfx1250) — compile-verified
//
#include <hip/hip_runtime.h>
#include <math.h>

typedef __attribute__((ext_vector_type(2))) float v2f;
typedef __attribute__((ext_vector_type(8))) float v8f;

// ---------------------------------------------------------------------------
// Utility kernels
// ---------------------------------------------------------------------------
__global__ void GCN_zero_kernel(float* __restrict__ p, int n) {
    int i = blockIdx.x * blockDim.x + threadIdx.x;
    if (i < n) p[i] = 0.0f;
}

// deg[dst] += 1 for every edge (segment_sum of ones at dst)
__global__ void GCN_degree_kernel(const int* __restrict__ dst,
                                  float* __restrict__ deg, int E) {
    int e = blockIdx.x * blockDim.x + threadIdx.x;
    if (e < E) atomicAdd(&deg[dst[e]], 1.0f);
}

// in-place: deg -> dis = (deg + 1)^-1/2   (self-loop adds 1)
__global__ void GCN_rsqrt_kernel(float* __restrict__ deg, int N) {
    int i = blockIdx.x * blockDim.x + threadIdx.x;
    if (i < N) deg[i] = 1.0f / sqrtf(deg[i] + 1.0f);
}

// ---------------------------------------------------------------------------
// Dense node transform hw = h @ W via V_WMMA_F32_16X16X4_F32, fused with the
// self-loop term agg = dis^2 * hw. One wave = 16 nodes. All operand loads are
// branchless: addresses clamped in-bounds (finite garbage), and only B is
// zero-masked in padded K slots (A*0 = 0 removes A-side garbage; M/N padding
// is never stored). Stores take a wave-uniform fast path for full tiles.
//
// Layouts per CDNA5 ISA 7.12.2 (f32, 2 VGPRs for A and B):
//   A 16x4 : lanes 0-15 -> M=lane,   K = 0(v0)/1(v1); lanes 16-31 -> K = 2/3
//   B 4x16 : lanes 0-15 -> N=lane,   K = 0(v0)/1(v1); lanes 16-31 -> K = 2/3
//   D 16x16: lane<16 -> (M=r, N=lane); lane>=16 -> (M=r+8, N=lane-16)
// ---------------------------------------------------------------------------
template<int DI, int DO>
__global__ void GCN_node_mm_wmma(const float* __restrict__ h,
                                 const float* __restrict__ W,
                                 const float* __restrict__ dis,
                                 float* __restrict__ hw,
                                 float* __restrict__ agg, int N) {
    const int wave = threadIdx.x >> 5;
    const int lane = threadIdx.x & 31;
    const int base = blockIdx.x * ((blockDim.x >> 5) * 16) + wave * 16;
    if (base >= N) return;                 // uniform per-wave exit (EXEC full)

    const int half = lane >> 4;            // 0: lanes 0-15, 1: lanes 16-31
    const int l    = lane & 15;
    const int rowA = (base + l < N) ? (base + l) : (N - 1);   // clamp: finite
    const int lB   = (l < DO) ? l : (DO - 1);                 // clamp: finite

    const int k0  = half * 2;              // K base: 0 or 2
    const int k0a = (k0     < DI) ? k0     : (DI - 1);
    const int k0b = (k0 + 1 < DI) ? k0 + 1 : (DI - 1);
    const float m0a = (k0     < DI) ? 1.0f : 0.0f;
    const float m0b = (k0 + 1 < DI) ? 1.0f : 0.0f;

    v2f a0, b0;
    a0.x = h[rowA * DI + k0a];
    a0.y = h[rowA * DI + k0b];
    b0.x = W[k0a * DO + lB] * m0a;         // zero-mask padded K slots
    b0.y = W[k0b * DO + lB] * m0b;

    v8f c = {};
    c = __builtin_amdgcn_wmma_f32_16x16x4_f32(false, a0, false, b0,
                                              (short)0, c, false, false);
    if constexpr (DI > 4) {
        const int k1  = 4 + half * 2;      // K base: 4 or 6
        const int k1a = (k1     < DI) ? k1     : (DI - 1);
        const int k1b = (k1 + 1 < DI) ? k1 + 1 : (DI - 1);
        const float m1a = (k1     < DI) ? 1.0f : 0.0f;
        const float m1b = (k1 + 1 < DI) ? 1.0f : 0.0f;
        v2f a1, b1;
        a1.x = h[rowA * DI + k1a];
        a1.y = h[rowA * DI + k1b];
        b1.x = W[k1a * DO + lB] * m1a;
        b1.y = W[k1b * DO + lB] * m1b;
        c = __builtin_amdgcn_wmma_f32_16x16x4_f32(false, a1, false, b1,
                                                  (short)0, c, false, false);
    }

    if (l < DO) {
        const int row0 = base + 8 * half;  // this lane covers rows row0..row0+7
        if (base + 16 <= N) {
            // fast path (all but the last partial tile): unguarded stores
#pragma unroll
            for (int r = 0; r < 8; ++r) {
                const int node = row0 + r;
                const float v = c[r];
                const float s = dis[node];
                hw[node * DO + l]  = v;
                agg[node * DO + l] = s * s * v;   // self-loop: deg^-1 * hw
            }
        } else {
#pragma unroll
            for (int r = 0; r < 8; ++r) {
                const int node = row0 + r;
                if (node < N) {
                    const float v = c[r];
                    const float s = dis[node];
                    hw[node * DO + l]  = v;
                    agg[node * DO + l] = s * s * v;
                }
            }
        }
    }
}

// message passing: agg[dst] += dis[src]*dis[dst] * hw[src]
template<int D>
__global__ void GCN_edge_scatter(const int* __restrict__ src,
                                 const int* __restrict__ dst,
                                 const float* __restrict__ dis,
                                 const float* __restrict__ hw,
                                 float* __restrict__ agg, int E) {
    int e = blockIdx.x * blockDim.x + threadIdx.x;
    if (e >= E) return;
    const int s = src[e];
    const int t = dst[e];
    const float w = dis[s] * dis[t];
    float msg[D];
#pragma unroll
    for (int n = 0; n < D; ++n) msg[n] = w * hw[s * D + n];
#pragma unroll
    for (int n = 0; n < D; ++n) atomicAdd(&agg[t * D + n], msg[n]);
}

// h_out = act(agg + b)   (act: tanh for layers 0..5, identity for layer 6)
template<int D>
__global__ void GCN_combine(const float* __restrict__ agg,
                            const float* __restrict__ b,
                            float* __restrict__ hout, int N, int act) {
    int i = blockIdx.x * blockDim.x + threadIdx.x;
    if (i >= N) return;
#pragma unroll
    for (int n = 0; n < D; ++n) {
        const float v = agg[i * D + n] + b[n];
        hout[i * D + n] = act ? tanhf(v) : v;
    }
}

// out = h @ Wc^T + bc ; also emit h (second tuple element) after out
__global__ void GCN_classifier(const float* __restrict__ h,
                               const float* __restrict__ Wc,
                               const float* __restrict__ bc,
                               float* __restrict__ out, int N) {
    int i = blockIdx.x * blockDim.x + threadIdx.x;
    if (i >= N) return;
    const float h0 = h[2 * i], h1 = h[2 * i + 1];
    out[2 * i]     = h0 * Wc[0] + h1 * Wc[1] + bc[0];
    out[2 * i + 1] = h0 * Wc[2] + h1 * Wc[3] + bc[1];
    out[2 * N + 2 * i]     = h0;
    out[2 * N + 2 * i + 1] = h1;
}

// ---------------------------------------------------------------------------
template<int DI, int DO>
static void run_layer(const float* hin, const float* W, const float* b,
                      const float* dis, float* hw, float* agg, float* hout,
                      const int* src, const int* dst, int N, int E, int act,
                      hipStream_t stream) {
    const int T   = 256;
    const int gN  = (N + T - 1) / T;
    const int gE  = (E + T - 1) / T;
    const int gMM = (N + 127) / 128;       // 8 waves/block * 16 nodes/wave

    GCN_node_mm_wmma<DI, DO><<<gMM, T, 0, stream>>>(hin, W, dis, hw, agg, N);
    GCN_edge_scatter<DO><<<gE, T, 0, stream>>>(src, dst, dis, hw, agg, E);
    GCN_combine<DO><<<gN, T, 0, stream>>>(agg, b, hout, N, act);
}

extern "C" void kernel_launch(void* const* d_in, const int* in_sizes, int n_in,
                              void* d_out, int out_size, void* d_ws, size_t ws_size,
                              hipStream_t stream) {
    const float* x  = (const float*)d_in[0];
    const int*   ei = (const int*)d_in[1];
    const int N = in_sizes[0] / 2;
    const int E = in_sizes[1] / 2;
    const int* src = ei;
    const int* dst = ei + E;

    // Locate the 7 W/b arrays + classifier weights; support both harness
    // flattenings of the Python lists (concatenated buffer vs. per-element).
    const float *Wp[7], *bp[7], *Wc, *bc;
    if (n_in >= 18) {
        for (int i = 0; i < 7; ++i) { Wp[i] = (const float*)d_in[2 + i];
                                      bp[i] = (const float*)d_in[9 + i]; }
        Wc = (const float*)d_in[16];
        bc = (const float*)d_in[17];
    } else {
        static const int wOff[7] = {0, 6, 24, 42, 51, 60, 66};
        static const int bOff[7] = {0, 3, 9, 12, 15, 18, 20};
        const float* Ws = (const float*)d_in[2];
        const float* bs = (const float*)d_in[3];
        for (int i = 0; i < 7; ++i) { Wp[i] = Ws + wOff[i]; bp[i] = bs + bOff[i]; }
        Wc = (const float*)d_in[4];
        bc = (const float*)d_in[5];
    }

    // Workspace layout (floats): dis[N] | hw[N*6] | agg[N*6] | hA[N*6] | hB[N*6]
    float* ws  = (float*)d_ws;
    float* dis = ws;
    float* hw  = dis + N;
    float* agg = hw + (size_t)N * 6;
    float* hA  = agg + (size_t)N * 6;
    float* hB  = hA + (size_t)N * 6;

    const int T  = 256;
    const int gN = (N + T - 1) / T;
    const int gE = (E + T - 1) / T;

    // Degree -> dis = (deg+1)^-1/2
    GCN_zero_kernel<<<gN, T, 0, stream>>>(dis, N);
    GCN_degree_kernel<<<gE, T, 0, stream>>>(dst, dis, E);
    GCN_rsqrt_kernel<<<gN, T, 0, stream>>>(dis, N);

    // 7 GCN layers, dims: 2->3->6->3->3->3->2->2 (tanh on layers 0..5)
    run_layer<2, 3>(x,  Wp[0], bp[0], dis, hw, agg, hA, src, dst, N, E, 1, stream);
    run_layer<3, 6>(hA, Wp[1], bp[1], dis, hw, agg, hB, src, dst, N, E, 1, stream);
    run_layer<6, 3>(hB, Wp[2], bp[2], dis, hw, agg, hA, src, dst, N, E, 1, stream);
    run_layer<3, 3>(hA, Wp[3], bp[3], dis, hw, agg, hB, src, dst, N, E, 1, stream);
    run_layer<3, 3>(hB, Wp[4], bp[4], dis, hw, agg, hA, src, dst, N, E, 1, stream);
    run_layer<3, 2>(hA, Wp[5], bp[5], dis, hw, agg, hB, src, dst, N, E, 1, stream);
    run_layer<2, 2>(hB, Wp[6], bp[6], dis, hw, agg, hA, src, dst, N, E, 0, stream);

    GCN_classifier<<<gN, T, 0, stream>>>(hA, Wc, bc, (float*)d_out, N);

    (void)out_size; (void)ws_size; (void)n_in;
}